// Seq2SeqModel_5729486373145
// MI455X (gfx1250) — compile-verified
//
#include <hip/hip_runtime.h>
#include <hip/hip_bf16.h>

// Problem constants (match reference)
#define BB   1024
#define TT   512
#define DD   64
#define HH   8
#define LL   3
#define G4H  32   // 4*H gates

typedef float v2f __attribute__((ext_vector_type(2)));
typedef float v8f __attribute__((ext_vector_type(8)));

// ---------------------------------------------------------------------------
// Phase 1: layer-0 input projection for encoder AND decoder via fp32 WMMA.
//   pre[b*T + t][n] = sum_k X[b*T+t][k] * Wih0[n][k]  + (bih0[n] + bhh0[n])
// One wave computes one 16x16 output tile. K=64 -> 16 x V_WMMA_F32_16X16X4_F32.
// Block = 128 threads = 4 waves: slot0/1 = enc (N-half 0/1), slot2/3 = dec.
// All 4 waves share the same 16 X rows -> L0 reuse.
// ---------------------------------------------------------------------------
__global__ __launch_bounds__(128) void lstm_pre_gemm(
    const float* __restrict__ X,        // [B*T, 64]
    const float* __restrict__ Wenc,     // [32, 64]
    const float* __restrict__ bih_enc,  // [3, 32] (layer0 = first 32)
    const float* __restrict__ bhh_enc,
    const float* __restrict__ Wdec,
    const float* __restrict__ bih_dec,
    const float* __restrict__ bhh_dec,
    float* __restrict__ pre_enc,        // [B*T, 32]
    float* __restrict__ pre_dec)
{
  const int lane  = threadIdx.x & 31;
  const int slot  = threadIdx.x >> 5;      // 0..3
  const int mtile = blockIdx.x;            // 0 .. B*T/16-1
  const bool enc  = (slot < 2);
  const int ntile = slot & 1;

  const int half = lane >> 4;              // 0/1: K split across lane halves
  const int m    = lane & 15;              // row within tile / col within N-half
  const int n    = ntile * 16 + m;         // gate index 0..31

  const float* W   = enc ? Wenc : Wdec;
  const float* bih = enc ? bih_enc : bih_dec;
  const float* bhh = enc ? bhh_enc : bhh_dec;
  float* out       = enc ? pre_enc : pre_dec;

  // Preload B-matrix fragments: b[v] = B[k = 4*kk + 2*half + v][n] = W[n][k]
  const float* wr = W + n * DD + 2 * half;
  v2f wb[16];
#pragma unroll
  for (int kk = 0; kk < 16; ++kk) {
    wb[kk].x = wr[4 * kk + 0];
    wb[kk].y = wr[4 * kk + 1];
  }

  const int r0 = mtile * 16;
  const float* xr = X + (size_t)(r0 + m) * DD + 2 * half;

  v8f acc = {0.f, 0.f, 0.f, 0.f, 0.f, 0.f, 0.f, 0.f};
#pragma unroll
  for (int kk = 0; kk < 16; ++kk) {
    v2f a;
    a.x = xr[4 * kk + 0];                  // A[m][4kk + 2*half + 0]
    a.y = xr[4 * kk + 1];                  // A[m][4kk + 2*half + 1]
    acc = __builtin_amdgcn_wmma_f32_16x16x4_f32(
        false, a, false, wb[kk], (short)0, acc, false, false);
  }

  const float bias = bih[n] + bhh[n];
  // D layout: vgpr v, lanes 0-15 -> row v, lanes 16-31 -> row v+8; col = n
#pragma unroll
  for (int v = 0; v < 8; ++v) {
    const int row = r0 + v + 8 * half;
    out[(size_t)row * G4H + n] = acc[v] + bias;
  }
}

// ---------------------------------------------------------------------------
// Phase 2: sequential scan. One wave == one batch item; lane == gate (0..31).
// h is wave-uniform -> recurrent dots via v_readlane (SGPR broadcast).
// ---------------------------------------------------------------------------
__device__ __forceinline__ float rl(float v, int l) {
  return __int_as_float(__builtin_amdgcn_readlane(__float_as_int(v), l));
}

__device__ __forceinline__ float sig_f(float x) {
  // 1 / (1 + 2^(-x*log2(e)))
  return __builtin_amdgcn_rcpf(1.0f + __builtin_amdgcn_exp2f(-1.44269504089f * x));
}

__device__ __forceinline__ float tanh_f(float x) {
#if __has_builtin(__builtin_amdgcn_tanhf)
  return __builtin_amdgcn_tanhf(x);
#else
  // tanh(x) = 2/(1 + 2^(-2x*log2(e))) - 1
  return __builtin_amdgcn_rcpf(1.0f + __builtin_amdgcn_exp2f(-2.88539008178f * x)) * 2.0f - 1.0f;
#endif
}

// sum_k w[k] * readlane(v, k)   (two accumulators to shorten the FMA chain)
__device__ __forceinline__ float dot8(const float (&w)[8], float v) {
  float a = 0.f, b = 0.f;
  a = fmaf(w[0], rl(v, 0), a);  b = fmaf(w[1], rl(v, 1), b);
  a = fmaf(w[2], rl(v, 2), a);  b = fmaf(w[3], rl(v, 3), b);
  a = fmaf(w[4], rl(v, 4), a);  b = fmaf(w[5], rl(v, 5), b);
  a = fmaf(w[6], rl(v, 6), a);  b = fmaf(w[7], rl(v, 7), b);
  return a + b;
}

// gates 'g' live one per lane (torch order i,f,g,o in lane groups of 8).
// Valid h/c state lives in lanes 0..7 (column j = lane).
__device__ __forceinline__ void cell_update(float g, int grp, float& h, float& c) {
  const float a  = (grp == 2) ? tanh_f(g) : sig_f(g); // group 2 = "g" gate -> tanh
  const float af = __shfl_down(a, 8, 32);   // sigmoid(f) into lanes 0..7
  const float ag = __shfl_down(a, 16, 32);  // tanh(g)
  const float ao = __shfl_down(a, 24, 32);  // sigmoid(o)
  c = fmaf(af, c, a * ag);                  // a == sigmoid(i) in lanes 0..7
  h = ao * tanh_f(c);
}

__device__ __forceinline__ void lstm_phase(
    const float* __restrict__ pre,       // [B*T, 32], layer-0 proj (+bias0)
    const float* __restrict__ Wih_rest,  // [2, 32, 8]
    const float* __restrict__ Whh,       // [3, 32, 8]
    const float* __restrict__ bih,       // [3, 32]
    const float* __restrict__ bhh,       // [3, 32]
    int b, int lane, int grp,
    float (&h)[LL], float (&c)[LL])
{
  float whh0[8], wih1[8], whh1[8], wih2[8], whh2[8];
#pragma unroll
  for (int k = 0; k < 8; ++k) {
    whh0[k] = Whh[lane * 8 + k];                 // layer 0
    wih1[k] = Wih_rest[lane * 8 + k];            // layer 1
    whh1[k] = Whh[256 + lane * 8 + k];
    wih2[k] = Wih_rest[256 + lane * 8 + k];      // layer 2
    whh2[k] = Whh[512 + lane * 8 + k];
  }
  const float b1 = bih[32 + lane] + bhh[32 + lane];
  const float b2 = bih[64 + lane] + bhh[64 + lane];

  const float* p = pre + (size_t)b * TT * G4H + lane;
  float pcur = p[0];

#pragma clang loop unroll(disable)
  for (int t = 0; t < TT; ++t) {
    // prefetch next timestep's input projection early
    float pnxt = (t + 1 < TT) ? p[(t + 1) * G4H] : 0.f;

    // layer 0: gates = pre (has x-proj + bias0) + Whh0 . h[0]
    float g = pcur + dot8(whh0, h[0]);
    cell_update(g, grp, h[0], c[0]);

    // layer 1: input = h[0]
    g = b1 + dot8(wih1, h[0]) + dot8(whh1, h[1]);
    cell_update(g, grp, h[1], c[1]);

    // layer 2: input = h[1]
    g = b2 + dot8(wih2, h[1]) + dot8(whh2, h[2]);
    cell_update(g, grp, h[2], c[2]);

    pcur = pnxt;
  }
}

__global__ __launch_bounds__(256) void lstm_recurrence(
    const float* __restrict__ pre_enc, const float* __restrict__ pre_dec,
    const float* __restrict__ h0, const float* __restrict__ c0,
    const float* __restrict__ eWih_rest, const float* __restrict__ eWhh,
    const float* __restrict__ ebih, const float* __restrict__ ebhh,
    const float* __restrict__ dWih_rest, const float* __restrict__ dWhh,
    const float* __restrict__ dbih, const float* __restrict__ dbhh,
    const float* __restrict__ linW,   // [1, 8]
    const float* __restrict__ linb,   // [1]
    float* __restrict__ out)          // [B, 1]
{
  const int lane = threadIdx.x & 31;
  const int grp  = lane >> 3;
  const int b    = blockIdx.x * (blockDim.x >> 5) + (threadIdx.x >> 5);

  float h[LL], c[LL];
  const int j = lane & 7;  // lanes 0..7 hold the real state; others replicate
#pragma unroll
  for (int l = 0; l < LL; ++l) {
    h[l] = h0[(size_t)l * BB * HH + (size_t)b * HH + j];
    c[l] = c0[(size_t)l * BB * HH + (size_t)b * HH + j];
  }

  // Encoder scan, then decoder scan continuing from the encoder state.
  lstm_phase(pre_enc, eWih_rest, eWhh, ebih, ebhh, b, lane, grp, h, c);
  lstm_phase(pre_dec, dWih_rest, dWhh, dbih, dbhh, b, lane, grp, h, c);

  // output head: pred[b] = lin_b + sum_j linW[j] * h[2][j]
  float acc = linb[0];
#pragma unroll
  for (int k = 0; k < 8; ++k) acc = fmaf(linW[k], rl(h[2], k), acc);
  if (lane == 0) out[b] = acc;
}

// ---------------------------------------------------------------------------
extern "C" void kernel_launch(void* const* d_in, const int* in_sizes, int n_in,
                              void* d_out, int out_size, void* d_ws, size_t ws_size,
                              hipStream_t stream) {
  const float* X            = (const float*)d_in[0];   // [B,T,64]
  const float* h0           = (const float*)d_in[1];   // [3,B,8]
  const float* c0           = (const float*)d_in[2];
  const float* enc_Wih0     = (const float*)d_in[3];   // [32,64]
  const float* enc_Wih_rest = (const float*)d_in[4];   // [2,32,8]
  const float* enc_Whh      = (const float*)d_in[5];   // [3,32,8]
  const float* enc_bih      = (const float*)d_in[6];   // [3,32]
  const float* enc_bhh      = (const float*)d_in[7];
  const float* dec_Wih0     = (const float*)d_in[8];
  const float* dec_Wih_rest = (const float*)d_in[9];
  const float* dec_Whh      = (const float*)d_in[10];
  const float* dec_bih      = (const float*)d_in[11];
  const float* dec_bhh      = (const float*)d_in[12];
  const float* linW         = (const float*)d_in[13];  // [1,8]
  const float* linb         = (const float*)d_in[14];  // [1]

  float* pre_enc = (float*)d_ws;                        // [B*T, 32]
  float* pre_dec = pre_enc + (size_t)BB * TT * G4H;     // [B*T, 32]

  // Phase 1: WMMA input projections (enc + dec fused in one launch)
  dim3 g1(BB * TT / 16), t1(128);
  lstm_pre_gemm<<<g1, t1, 0, stream>>>(X, enc_Wih0, enc_bih, enc_bhh,
                                       dec_Wih0, dec_bih, dec_bhh,
                                       pre_enc, pre_dec);

  // Phase 2: one wave per batch item, 8 waves per block
  dim3 g2(BB / 8), t2(256);
  lstm_recurrence<<<g2, t2, 0, stream>>>(pre_enc, pre_dec, h0, c0,
                                         enc_Wih_rest, enc_Whh, enc_bih, enc_bhh,
                                         dec_Wih_rest, dec_Whh, dec_bih, dec_bhh,
                                         linW, linb, (float*)d_out);
}